// GCNEncoder_61881888801355
// MI455X (gfx1250) — compile-verified
//
#include <hip/hip_runtime.h>

// ---------------------------------------------------------------------------
// GCNEncoder for MI455X (gfx1250, wave32)
//   out = PReLU( D^-1/2 (A+I) D^-1/2 (x @ W) + bias )
// Folded form: s = dinv * (x@W);  out[c] = dinv[c]*(sum_e s[row_e] + s[c]) + b
// ---------------------------------------------------------------------------

typedef __attribute__((ext_vector_type(2))) float v2f;
typedef __attribute__((ext_vector_type(4))) float v4f;
typedef __attribute__((ext_vector_type(8))) float v8f;

#define N_NODES 100000
#define N_EDGES 1200000
#define IN_DIM  256
#define Z_DIM   64
#define M_TILES (N_NODES / 16)     // 6250, exact
#define LDS_STRIDE 72              // pad: 16B-aligned rows, lane-halves hit disjoint banks

// ---------------- degree / normalization ----------------

__global__ __launch_bounds__(256) void zero_counts_kernel(unsigned* __restrict__ cnt) {
    int i = blockIdx.x * 256 + threadIdx.x;
    if (i < N_NODES) cnt[i] = 0u;
}

__global__ __launch_bounds__(256) void count_deg_kernel(const int* __restrict__ ei,
                                                        unsigned* __restrict__ cnt) {
    int e = blockIdx.x * 256 + threadIdx.x;
    if (e < N_EDGES) {
        int col = ei[N_EDGES + e];          // destination node
        atomicAdd(&cnt[col], 1u);
    }
}

__global__ __launch_bounds__(256) void dinv_kernel(const unsigned* __restrict__ cnt,
                                                   float* __restrict__ dinv) {
    int i = blockIdx.x * 256 + threadIdx.x;
    if (i < N_NODES) {
        float d = (float)cnt[i] + 1.0f;     // +1 self loop; always > 0
        dinv[i] = __frsqrt_rn(d);
    }
}

// ---------------- GEMM: s = dinv * (x @ W), out seeded with s (self loop) ----

__global__ __launch_bounds__(256) void gemm_scale_kernel(const float* __restrict__ x,
                                                         const float* __restrict__ W,
                                                         const float* __restrict__ dinv,
                                                         float* __restrict__ s,
                                                         float* __restrict__ out) {
    __shared__ float Wl[IN_DIM * LDS_STRIDE];

    const int tid = threadIdx.x;

    // Stage W (256x64 f32, 64KB) into LDS once per workgroup, float4 chunks.
    for (int i = tid; i < (IN_DIM * Z_DIM) / 4; i += 256) {
        int r  = i >> 4;                  // 16 float4 per row of 64
        int c4 = (i & 15) << 2;
        v4f w = *(const v4f*)(W + r * Z_DIM + c4);
        *(v4f*)(&Wl[r * LDS_STRIDE + c4]) = w;
    }
    __syncthreads();

    const int wave = tid >> 5;
    const int lane = tid & 31;
    const int mt   = blockIdx.x * 8 + wave;   // one 16-row tile per wave
    if (mt >= M_TILES) return;                // wave-uniform: EXEC stays all-ones

    const int m0   = mt * 16;
    const int half = lane >> 4;               // K-half select (A & B fragments)
    const int l16  = lane & 15;               // M-row (A) / N-col (B,C)

    v8f z = {};
    v8f acc0 = z, acc1 = z, acc2 = z, acc3 = z;

    const float* xrow = x + (size_t)(m0 + l16) * IN_DIM + 2 * half;

    #pragma unroll 4
    for (int k = 0; k < IN_DIM; k += 4) {
        // A 16x4 f32 fragment: lanes 0-15 K={k,k+1}, lanes 16-31 K={k+2,k+3}
        v2f a = *(const v2f*)(xrow + k);
        const int kr = k + 2 * half;
        // B 4x16 f32 fragments for the four 16-col tiles
        v2f b0, b1, b2, b3;
        b0.x = Wl[(kr    ) * LDS_STRIDE +  0 + l16];
        b0.y = Wl[(kr + 1) * LDS_STRIDE +  0 + l16];
        b1.x = Wl[(kr    ) * LDS_STRIDE + 16 + l16];
        b1.y = Wl[(kr + 1) * LDS_STRIDE + 16 + l16];
        b2.x = Wl[(kr    ) * LDS_STRIDE + 32 + l16];
        b2.y = Wl[(kr + 1) * LDS_STRIDE + 32 + l16];
        b3.x = Wl[(kr    ) * LDS_STRIDE + 48 + l16];
        b3.y = Wl[(kr + 1) * LDS_STRIDE + 48 + l16];
        acc0 = __builtin_amdgcn_wmma_f32_16x16x4_f32(false, a, false, b0, (short)0, acc0, false, false);
        acc1 = __builtin_amdgcn_wmma_f32_16x16x4_f32(false, a, false, b1, (short)0, acc1, false, false);
        acc2 = __builtin_amdgcn_wmma_f32_16x16x4_f32(false, a, false, b2, (short)0, acc2, false, false);
        acc3 = __builtin_amdgcn_wmma_f32_16x16x4_f32(false, a, false, b3, (short)0, acc3, false, false);
    }

    // dinv for the 8 rows this lane holds (C layout: row = m0 + half*8 + r)
    float dv[8];
    #pragma unroll
    for (int r = 0; r < 8; ++r) dv[r] = dinv[m0 + half * 8 + r];

    v8f accs[4] = {acc0, acc1, acc2, acc3};
    #pragma unroll
    for (int nt = 0; nt < 4; ++nt) {
        #pragma unroll
        for (int r = 0; r < 8; ++r) {
            float v = accs[nt][r] * dv[r];
            size_t off = (size_t)(m0 + half * 8 + r) * Z_DIM + nt * 16 + l16;
            s[off]   = v;   // pre-scaled messages
            out[off] = v;   // accumulator seeded with self-loop term
        }
    }
}

// ---------------- edge scatter: out[col] += s[row] ----------------

__global__ __launch_bounds__(256) void scatter_kernel(const int* __restrict__ ei,
                                                      const float* __restrict__ s,
                                                      float* __restrict__ out) {
    unsigned gid = blockIdx.x * 256u + threadIdx.x;   // (edge, channel) pair
    int e = (int)(gid >> 6);
    int k = (int)(gid & 63u);
    if (e >= N_EDGES) return;
    int row = ei[e];
    int col = ei[N_EDGES + e];
    float v = s[(size_t)row * Z_DIM + k];
    unsafeAtomicAdd(&out[(size_t)col * Z_DIM + k], v);   // native global_atomic_add_f32
}

// ---------------- finalize: out = PReLU(dinv*out + bias) ----------------

__global__ __launch_bounds__(256) void finalize_kernel(float* __restrict__ out,
                                                       const float* __restrict__ dinv,
                                                       const float* __restrict__ bias,
                                                       const float* __restrict__ pa) {
    int gid = blockIdx.x * 256 + threadIdx.x;
    if (gid >= N_NODES * Z_DIM) return;
    int k = gid & 63;
    float v = dinv[gid >> 6] * out[gid] + bias[k];
    out[gid] = (v >= 0.0f) ? v : pa[k] * v;
}

// ---------------- launch ----------------

extern "C" void kernel_launch(void* const* d_in, const int* in_sizes, int n_in,
                              void* d_out, int out_size, void* d_ws, size_t ws_size,
                              hipStream_t stream) {
    const float* x    = (const float*)d_in[0];
    const int*   ei   = (const int*)d_in[1];   // [2, N_EDGES] flat (row, then col)
    // d_in[2] = edge_type: unused by the reference forward
    const float* W    = (const float*)d_in[3];
    const float* bias = (const float*)d_in[4];
    const float* pa   = (const float*)d_in[5];
    float* out = (float*)d_out;

    // Workspace layout (all ws re-initialized every call; no cross-call state):
    //   [0, 400000)        : unsigned cnt[N_NODES]
    //   [400000, 800000)   : float dinv[N_NODES]
    //   [800000, ...)      : float s[N_NODES * Z_DIM]   (25.6 MB, 16B aligned)
    unsigned* cnt = (unsigned*)d_ws;
    float* dinv   = (float*)((char*)d_ws + (size_t)N_NODES * 4);
    float* s      = (float*)((char*)d_ws + (size_t)2 * N_NODES * 4);

    zero_counts_kernel<<<(N_NODES + 255) / 256, 256, 0, stream>>>(cnt);
    count_deg_kernel<<<(N_EDGES + 255) / 256, 256, 0, stream>>>(ei, cnt);
    dinv_kernel<<<(N_NODES + 255) / 256, 256, 0, stream>>>(cnt, dinv);

    gemm_scale_kernel<<<(M_TILES + 7) / 8, 256, 0, stream>>>(x, W, dinv, s, out);

    const long long pairs = (long long)N_EDGES * Z_DIM;     // 76.8M atomics, L2-resident
    scatter_kernel<<<(unsigned)((pairs + 255) / 256), 256, 0, stream>>>(ei, s, out);

    finalize_kernel<<<(N_NODES * Z_DIM + 255) / 256, 256, 0, stream>>>(out, dinv, bias, pa);
}